// AFMADEBlock_33466385170819
// MI455X (gfx1250) — compile-verified
//
#include <hip/hip_runtime.h>
#include <math.h>

// ---------------------------------------------------------------------------
// MADE inverse-autoregressive scan, restructured incrementally for MI455X.
//  B=128 batch rows -> 8 independent workgroups of 16 rows (wave32, 8 waves).
//  Per step:
//    A: rank-1 z0 update with new y column (z0 in LDS, [u][net][b] layout so
//       the update is contiguous float4 = ds_*_b128)
//    B: finalize <=3 hidden-0 units (degree==i), scatter into z1 with
//       V_WMMA_F32_16X16X4_F32. z1 accumulators are REGISTER-RESIDENT:
//       each wave persistently owns 8 (net, v-tile) tiles (8 x v8f VGPRs).
//    C (fused): masked Wo-row dot over branchless elu(z1+b1); per-lane
//       accumulation across tiles, then ONE 16-lane xor-shuffle tree per row.
//    D: reduce per-wave partials, form y column, running logstd sum.
// ---------------------------------------------------------------------------

constexpr int   B   = 128;
constexpr int   D   = 256;
constexpr int   H   = 512;
constexpr int   BT  = 16;   // batch rows per workgroup
constexpr int   NT  = 256;  // threads per workgroup (8 x wave32)
constexpr float EPS = 1e-12f;

typedef __attribute__((ext_vector_type(2))) float v2f;
typedef __attribute__((ext_vector_type(8))) float v8f;

#define DEV __device__ __forceinline__

DEV int degH(int u) { return (u % (D - 1)) + 1; }   // hidden degrees 1..255

// Branchless ELU: v_min + v_exp_f32 + v_cndmask, no EXEC manipulation.
DEV float eluf(float z) {
  float e = __expf(fminf(z, 0.f)) - 1.f;
  return z > 0.f ? z : e;
}

// ---- workspace layout (mask-baked / transposed weights), in floats ----------
constexpr int W0T_SZ  = D * H;     // W0^T masked: [D][H]
constexpr int W1T_SZ  = H * H;     // W1^T masked: [H][H]  (row u = column u of W1)
constexpr int WOM_SZ  = D * H;     // Wo masked:   [D][H]
constexpr int OFF_W0M = 0;
constexpr int OFF_W0L = OFF_W0M + W0T_SZ;
constexpr int OFF_W1M = OFF_W0L + W0T_SZ;
constexpr int OFF_W1L = OFF_W1M + W1T_SZ;
constexpr int OFF_WOM = OFF_W1L + W1T_SZ;
constexpr int OFF_WOL = OFF_WOM + WOM_SZ;
constexpr int WS_FLOATS = OFF_WOL + WOM_SZ;   // 1,048,576 floats = 4 MB

// ---- dynamic LDS layout (floats) -------------------------------------------
constexpr int SMEM_FLOATS =
    2*BT*H      // z0  [u][net][b]   (interleaved for float4 updates)
  + 2*H         // b0s
  + 2*H         // b1s
  + 2*4*BT      // aA    A-fragments: h0 of up-to-4 finalized units
  + 2*4*H       // w1s   staged W1^T rows for those units
  + 2*H         // wos   staged masked Wo row i
  + 2*4*BT      // partsC per-wave output partials [net][wgrp][M]
  + BT          // ycol
  + BT;         // lsum

// ---------------------------------------------------------------------------
// Prep: bake masks + transpose W0, W1 into d_ws; mask-copy Wo.
// ---------------------------------------------------------------------------
__global__ void prep_weights(const float* __restrict__ W0m, const float* __restrict__ W1m,
                             const float* __restrict__ Wom, const float* __restrict__ W0l,
                             const float* __restrict__ W1l, const float* __restrict__ Wol,
                             float* __restrict__ ws)
{
  float* t0m = ws + OFF_W0M;  float* t0l = ws + OFF_W0L;
  float* t1m = ws + OFF_W1M;  float* t1l = ws + OFF_W1L;
  float* tom = ws + OFF_WOM;  float* tol = ws + OFF_WOL;
  const int gsz = gridDim.x * blockDim.x;
  const int g0  = blockIdx.x * blockDim.x + threadIdx.x;

  for (int k = g0; k < D*H; k += gsz) {
    int j = k / H, u = k % H;                       // j: input col / output row; u: hidden unit
    float m0 = (degH(u) >= j + 1) ? 1.f : 0.f;      // m0[u,j] = dh[u] >= din[j]
    t0m[k] = W0m[u*D + j] * m0;
    t0l[k] = W0l[u*D + j] * m0;
    float mo = (degH(u) <= j) ? 1.f : 0.f;          // mo[j,u] = din[j] > dh[u]
    tom[k] = Wom[k] * mo;
    tol[k] = Wol[k] * mo;
  }
  for (int k = g0; k < H*H; k += gsz) {
    int u = k / H, v = k % H;                       // t1[u][v] = W1[v][u] * m1[v,u]
    float m1 = (degH(v) >= degH(u)) ? 1.f : 0.f;
    t1m[k] = W1m[v*H + u] * m1;
    t1l[k] = W1l[v*H + u] * m1;
  }
}

// ---------------------------------------------------------------------------
// Main persistent per-batch-tile scan. TR = weights are pre-masked/transposed.
// ---------------------------------------------------------------------------
template<bool TR>
__global__ __launch_bounds__(NT, 1)
void made_scan(const float* __restrict__ x,
               const float* __restrict__ W0m, const float* __restrict__ b0m,
               const float* __restrict__ W1m, const float* __restrict__ b1m,
               const float* __restrict__ Wom, const float* __restrict__ bom,
               const float* __restrict__ W0l, const float* __restrict__ b0l,
               const float* __restrict__ W1l, const float* __restrict__ b1l,
               const float* __restrict__ Wol, const float* __restrict__ bol,
               float* __restrict__ out)
{
  extern __shared__ float smem[];
  float* z0     = smem;                 // [H][2][BT]  element (net,b,u) -> (u*2+net)*BT+b
  float* b0s    = z0 + 2*BT*H;          // [2][H]
  float* b1s    = b0s + 2*H;            // [2][H]
  float* aA     = b1s + 2*H;            // [2][4][BT]
  float* w1s    = aA + 2*4*BT;          // [2][4][H]
  float* wos    = w1s + 2*4*H;          // [2][H]
  float* partsC = wos + 2*H;            // [2][4][BT]  (net, wave-group, M)
  float* ycol   = partsC + 2*4*BT;      // [BT]
  float* lsum   = ycol + BT;            // [BT]

  const int tid  = threadIdx.x;
  const int rb   = blockIdx.x * BT;
  const int lane = tid & 31;
  const int wave = tid >> 5;
  const int ln   = lane & 15;
  const int hi   = lane >> 4;           // 0: lanes 0-15, 1: lanes 16-31
  const int wnet = wave >> 2;           // this wave's network (0=mu, 1=lv)
  const int wgrp = wave & 3;            // this wave's v-tile group

  // Register-resident z1 accumulators: 8 tiles x v8f = 64 VGPRs/lane.
  v8f zacc[8];
  #pragma unroll
  for (int t = 0; t < 8; ++t) zacc[t] = (v8f)0.f;

  // ---- init recurrent state + bias staging ----
  for (int k = tid; k < 2*BT*H; k += NT) z0[k] = 0.f;
  for (int k = tid; k < H; k += NT) {
    b0s[k] = b0m[k];  b0s[H + k] = b0l[k];
    b1s[k] = b1m[k];  b1s[H + k] = b1l[k];
  }
  if (tid < BT) { ycol[tid] = 0.f; lsum[tid] = 0.f; }
  __syncthreads();

  for (int i = 0; i < D; ++i) {
    // ---- Phase A: z0 += y[:, i-1] (x) W0masked^T[i-1, :]  (both nets) ----
    if (i > 0) {
      float4 yv4[4];
      #pragma unroll
      for (int q = 0; q < 4; ++q) yv4[q] = ((const float4*)ycol)[q];
      for (int u = tid; u < H; u += NT) {
        float wm, wl;
        if (TR) {
          wm = W0m[(i - 1) * H + u];
          wl = W0l[(i - 1) * H + u];
        } else {
          float mk = (degH(u) >= i) ? 1.f : 0.f;
          wm = W0m[u * D + (i - 1)] * mk;
          wl = W0l[u * D + (i - 1)] * mk;
        }
        float4* zrow = (float4*)&z0[(u * 2) * BT];   // 8 float4: net0 b0..15, net1 b0..15
        #pragma unroll
        for (int q = 0; q < 4; ++q) {
          float4 zv = zrow[q];
          zv.x += yv4[q].x * wm;  zv.y += yv4[q].y * wm;
          zv.z += yv4[q].z * wm;  zv.w += yv4[q].w * wm;
          zrow[q] = zv;
        }
        #pragma unroll
        for (int q = 0; q < 4; ++q) {
          float4 zv = zrow[4 + q];
          zv.x += yv4[q].x * wl;  zv.y += yv4[q].y * wl;
          zv.z += yv4[q].z * wl;  zv.w += yv4[q].w * wl;
          zrow[4 + q] = zv;
        }
      }
    }
    __syncthreads();

    // ---- Phase B1: stage h0 of units with degree==i (u = i-1 + 255*s) ----
    if (tid < 2 * 4 * BT) {
      int b = tid & (BT - 1);
      int s = (tid >> 4) & 3;
      int net = tid >> 6;
      int u = (i - 1) + 255 * s;
      float h = 0.f;
      if (i >= 1 && s < 3 && u < H)
        h = eluf(z0[(u * 2 + net) * BT + b] + b0s[net * H + u]);
      aA[(net * 4 + s) * BT + b] = h;
    }
    // stage corresponding W1^T rows (zero-padded to K=4)
    for (int k = tid; k < 2 * 4 * H; k += NT) {
      int v = k & (H - 1);
      int s = (k >> 9) & 3;
      int net = k >> 11;
      int u = (i - 1) + 255 * s;
      float w = 0.f;
      if (i >= 1 && s < 3 && u < H) {
        if (TR) w = (net ? W1l : W1m)[u * H + v];
        else    w = (net ? W1l : W1m)[v * H + u] * ((degH(v) >= i) ? 1.f : 0.f);
      }
      w1s[k] = w;
    }
    // stage masked Wo row i
    for (int k = tid; k < 2 * H; k += NT) {
      int v = k & (H - 1);
      int net = k >> 9;
      float w = (net ? Wol : Wom)[i * H + v];
      if (!TR) w *= (degH(v) <= i) ? 1.f : 0.f;
      wos[k] = w;
    }
    // prefetch next step's W0^T row (global_prefetch_b8 path)
    if (TR && (i + 1) < D && tid < 64) {
      int net  = tid & 1;
      int line = tid >> 1;                       // 32 x 128B lines = 4 KB (2 KB row i + row i+1)
      const float* nb = (net ? W0l : W0m) + i * H;
      __builtin_prefetch(nb + line * 32, 0, 3);
    }
    __syncthreads();

    // ---- Phase B2 + fused C: WMMA scatter into register z1; per-lane masked
    //      output partials across tiles; ONE shuffle tree per row at the end.
    {
      const int k0 = hi ? 2 : 0;                 // A/B fragment: vgpr0=K{0|2}, vgpr1=K{1|3}
      v2f a;                                     // A fragment is tile-invariant
      a.x = aA[(wnet * 4 + k0    ) * BT + ln];
      a.y = aA[(wnet * 4 + k0 + 1) * BT + ln];

      float redM[8];                             // per-(M=mb+r) per-lane partials
      #pragma unroll
      for (int r = 0; r < 8; ++r) redM[r] = 0.f;

      #pragma unroll
      for (int t = 0; t < 8; ++t) {
        const int vt = wgrp * 8 + t;
        v2f bb;
        bb.x = w1s[(wnet * 4 + k0    ) * H + vt * 16 + ln];
        bb.y = w1s[(wnet * 4 + k0 + 1) * H + vt * 16 + ln];
        zacc[t] = __builtin_amdgcn_wmma_f32_16x16x4_f32(
                      false, a, false, bb, (short)0, zacc[t], false, false);

        // fused output contribution for held elements (M=mb+r, N=ln), v=vt*16+ln
        const float wv = wos[wnet * H + vt * 16 + ln];
        const float bv = b1s[wnet * H + vt * 16 + ln];
        #pragma unroll
        for (int r = 0; r < 8; ++r)
          redM[r] += wv * eluf(zacc[t][r] + bv);
      }
      // reduce over N (16 lanes within each half) once per row
      #pragma unroll
      for (int r = 0; r < 8; ++r) {
        float e = redM[r];
        e += __shfl_xor(e, 1, 16);
        e += __shfl_xor(e, 2, 16);
        e += __shfl_xor(e, 4, 16);
        e += __shfl_xor(e, 8, 16);
        redM[r] = e;                             // all lanes in half hold the sum
      }
      if (ln == 0) {                             // lane 0 -> M=0..7, lane 16 -> M=8..15
        const int mb = hi ? 8 : 0;
        #pragma unroll
        for (int r = 0; r < 8; ++r)
          partsC[(wnet * 4 + wgrp) * BT + (mb + r)] = redM[r];
      }
    }
    __syncthreads();

    // ---- Phase D: reduce per-wave partials, form y column, logstd sum ----
    if (tid < BT) {
      int b = tid;
      float sm = bom[i];
      float sl = bol[i];
      #pragma unroll
      for (int w = 0; w < 4; ++w) {
        sm += partsC[(0 * 4 + w) * BT + b];
        sl += partsC[(1 * 4 + w) * BT + b];
      }
      float ls = 0.5f * sl;
      lsum[b] += ls;
      float yv = (x[(rb + b) * D + i] - sm) / (__expf(ls) + EPS);
      ycol[b] = yv;
      out[(rb + b) * D + i] = yv;                // y output
      if (i == D - 1) out[B * D + rb + b] = lsum[b];   // logstd.sum(axis=1)
    }
    __syncthreads();
  }
}

// ---------------------------------------------------------------------------
extern "C" void kernel_launch(void* const* d_in, const int* in_sizes, int n_in,
                              void* d_out, int out_size, void* d_ws, size_t ws_size,
                              hipStream_t stream)
{
  (void)in_sizes; (void)n_in; (void)out_size;
  const float* x   = (const float*)d_in[0];
  const float* w0m = (const float*)d_in[1];  const float* b0m = (const float*)d_in[2];
  const float* w1m = (const float*)d_in[3];  const float* b1m = (const float*)d_in[4];
  const float* wom = (const float*)d_in[5];  const float* bom = (const float*)d_in[6];
  const float* w0l = (const float*)d_in[7];  const float* b0l = (const float*)d_in[8];
  const float* w1l = (const float*)d_in[9];  const float* b1l = (const float*)d_in[10];
  const float* wol = (const float*)d_in[11]; const float* bol = (const float*)d_in[12];
  float* out = (float*)d_out;

  const size_t shmem = (size_t)SMEM_FLOATS * sizeof(float);
  const size_t need  = (size_t)WS_FLOATS * sizeof(float);

  if (ws_size >= need) {
    float* ws = (float*)d_ws;
    prep_weights<<<512, 256, 0, stream>>>(w0m, w1m, wom, w0l, w1l, wol, ws);
    (void)hipFuncSetAttribute((const void*)made_scan<true>,
                              hipFuncAttributeMaxDynamicSharedMemorySize, (int)shmem);
    made_scan<true><<<B / BT, NT, shmem, stream>>>(
        x, ws + OFF_W0M, b0m, ws + OFF_W1M, b1m, ws + OFF_WOM, bom,
           ws + OFF_W0L, b0l, ws + OFF_W1L, b1l, ws + OFF_WOL, bol, out);
  } else {
    (void)hipFuncSetAttribute((const void*)made_scan<false>,
                              hipFuncAttributeMaxDynamicSharedMemorySize, (int)shmem);
    made_scan<false><<<B / BT, NT, shmem, stream>>>(
        x, w0m, b0m, w1m, b1m, wom, bom, w0l, b0l, w1l, b1l, wol, bol, out);
  }
}